// QuantumGate_65481071399129
// MI455X (gfx1250) — compile-verified
//
#include <hip/hip_runtime.h>
#include <math.h>

typedef __attribute__((ext_vector_type(2))) float v2f;
typedef __attribute__((ext_vector_type(8))) float v8f;

#define IN_F    512
#define NQ      10
#define KSTEPS  (IN_F / 4)        // 128 WMMA k-steps of K=4
#define WPB     4                 // waves per block
#define TPB     (WPB * 32)

// Closed form of the reference (RX layer + CNOT chain + <Z_k>):
//   out[m][k] = prod_{j<=k} cos( x[m]·W[j] + b[j] + theta[j] )
__global__ __launch_bounds__(TPB)
void qgate_kernel(const float* __restrict__ x,
                  const float* __restrict__ W,
                  const float* __restrict__ bias,
                  const float* __restrict__ theta,
                  float* __restrict__ out,
                  int batch)
{
    // W^T (padded to 16 N-columns) staged in LDS, pre-swizzled into the exact
    // V_WMMA_F32_16X16X4_F32 B-fragment layout:
    //   entry e = m*32 + lane holds float2:
    //     lane 0-15 : { W[n][4m+0], W[n][4m+1] }, n = lane      (VGPR0=K0, VGPR1=K1)
    //     lane 16-31: { W[n][4m+2], W[n][4m+3] }, n = lane-16   (VGPR0=K2, VGPR1=K3)
    __shared__ float lbs[KSTEPS * 32 * 2];      // 32 KB

    const int tid = threadIdx.x;
    for (int e = tid; e < KSTEPS * 32; e += TPB) {
        const int m  = e >> 5;
        const int ln = e & 31;
        const int nn = ln & 15;
        const int kb = 4 * m + ((ln >> 4) << 1);
        float vx = 0.f, vy = 0.f;
        if (nn < NQ) {
            const float* wp = W + nn * IN_F + kb;
            vx = wp[0];
            vy = wp[1];
        }
        lbs[2 * e + 0] = vx;
        lbs[2 * e + 1] = vy;
    }
    __syncthreads();

    const int lane = tid & 31;
    const int wave = tid >> 5;
    const int tile = blockIdx.x * WPB + wave;
    const int ntiles = batch >> 4;
    if (tile >= ntiles) return;          // wave-uniform: EXEC stays all-1s below

    const int rowbase = tile << 4;
    const int n  = lane & 15;            // N index (output column) in B/C layout
    const int hi = lane >> 4;

    // A-fragment source (ISA 16x4 f32 A layout): lane holds row rowbase+n,
    // K pair starting at 4m + 2*hi.
    const float* ap = x + (size_t)(rowbase + n) * IN_F + (hi << 1);
    const v2f* bfrag = (const v2f*)lbs;

    v8f acc0 = {}; v8f acc1 = {};        // two accumulators to break RAW chains
    #pragma unroll 4
    for (int m = 0; m < KSTEPS; m += 2) {
        if ((m & 7) == 0)                // uniform; ~256B ahead
            __builtin_prefetch(ap + 4 * (m + 16), 0, 1);
        v2f a0 = *(const v2f*)(ap + 4 * (m + 0));
        v2f b0 = bfrag[(m + 0) * 32 + lane];
        acc0 = __builtin_amdgcn_wmma_f32_16x16x4_f32(false, a0, false, b0,
                                                     (short)0, acc0, false, false);
        v2f a1 = *(const v2f*)(ap + 4 * (m + 1));
        v2f b1 = bfrag[(m + 1) * 32 + lane];
        acc1 = __builtin_amdgcn_wmma_f32_16x16x4_f32(false, a1, false, b1,
                                                     (short)0, acc1, false, false);
    }

    float bt = 0.f;
    if (n < NQ) bt = bias[n] + theta[n];  // padded columns: acc==0, bt==0 -> cos=1

    #pragma unroll
    for (int i = 0; i < 8; ++i) {
        // C layout: VGPR i, lanes 0-15 -> row i, lanes 16-31 -> row 8+i; N = lane&15
        float v = cosf(acc0[i] + acc1[i] + bt);
        // inclusive prefix product over N within each 16-lane segment
        #pragma unroll
        for (int off = 1; off < 16; off <<= 1) {
            float o = __shfl_up(v, off, 16);
            v = (n >= off) ? v * o : v;
        }
        if (n < NQ) {
            const int row = rowbase + (hi ? (8 + i) : i);
            out[(size_t)row * NQ + n] = v;
        }
    }
}

extern "C" void kernel_launch(void* const* d_in, const int* in_sizes, int n_in,
                              void* d_out, int out_size, void* d_ws, size_t ws_size,
                              hipStream_t stream)
{
    const float* x     = (const float*)d_in[0];
    const float* W     = (const float*)d_in[1];
    const float* b     = (const float*)d_in[2];
    const float* theta = (const float*)d_in[3];
    float* out = (float*)d_out;

    const int batch  = in_sizes[0] / IN_F;       // 16384
    const int ntiles = batch >> 4;               // 1024 tiles of 16 rows
    const int blocks = (ntiles + WPB - 1) / WPB; // 256 blocks x 4 waves
    qgate_kernel<<<blocks, TPB, 0, stream>>>(x, W, b, theta, out, batch);
}